// MSDABlock2_72928544686129
// MI455X (gfx1250) — compile-verified
//
#include <hip/hip_runtime.h>
#include <math.h>

// ---------------------------------------------------------------------------
// MSDA / NAFNet-style block for MI455X (gfx1250, wave32, WMMA).
// All 1x1 convs run as f32 WMMA GEMMs (V_WMMA_F32_16X16X4_F32): the whole
// block is HBM-bound (~20 FLOP/byte vs 23.3 TB/s), so full-precision f32
// matrix ops are free and bit-match the fp32 reference. Depthwise stages are
// the dominant byte movers -> 4 pixels/thread with b128 loads/stores.
// ---------------------------------------------------------------------------

typedef __attribute__((ext_vector_type(2))) float v2f;
typedef __attribute__((ext_vector_type(8))) float v8f;
typedef float f4a __attribute__((ext_vector_type(4), aligned(16)));  // aligned b128
typedef float f4u __attribute__((ext_vector_type(4), aligned(4)));   // 4B-aligned b128

static constexpr int Bb  = 4;
static constexpr int Cc  = 64;
static constexpr int DWc = 128;
static constexpr int Hh  = 256;
static constexpr int Ww  = 256;
static constexpr int HW  = Hh * Ww;           // 65536 = 1<<16
static constexpr long NPIX = (long)Bb * HW;   // 262144

// ---------------------------------------------------------------------------
// Fused (optional LayerNorm) + 1x1 conv GEMM:  out[M, pix] = W[M,64] x X[64, pix]
// Block = 256 threads = 8 wave32s. Each wave computes a 16x16 tile with
// 16 chained V_WMMA_F32_16X16X4_F32 (K = 64 = 16 steps of 4).
// EPI: 0 = bias only; 1 = x*(sca + (acc+bias)) [SCA+LKA combine];
//      2 = e0 + e1[m]*(acc+bias)              [residual with beta/gamma]
// ---------------------------------------------------------------------------
template <int M, int EPI, bool DO_LN>
__global__ __launch_bounds__(256) void gemm1x1(
    const float* __restrict__ X,  const float* __restrict__ Wg,
    const float* __restrict__ bias,
    const float* __restrict__ ln_w, const float* __restrict__ ln_b,
    const float* __restrict__ e0, const float* __restrict__ e1,
    float* __restrict__ out)
{
    constexpr int K   = 64;
    constexpr int MT  = M / 16;      // M-tiles (waves along M)
    constexpr int NWT = 8 / MT;      // N-tiles per block
    constexpr int NT  = NWT * 16;    // pixels per block
    constexpr int QT  = NT / 4;      // float4 quads per row of the X tile

    __shared__ float sW[M * (K + 1)];     // +1 pad: A-frag column reads conflict-free
    __shared__ float sX[K * (NT + 1)];
    __shared__ float sMu[NT], sRs[NT];

    const int  tid = threadIdx.x;
    const long n0  = (long)blockIdx.x * NT;       // HW % NT == 0 -> tile within one image
    const int  b   = (int)(n0 >> 16);             // n0 / HW
    const int  p0  = (int)(n0 & (HW - 1));

    // Cooperative load (b128 from global, scalar into padded LDS pitch).
    for (int i = tid; i < M * K / 4; i += 256) {
        const f4a v = *(const f4a*)(Wg + 4 * i);
        const int r = (4 * i) / K, cidx = (4 * i) % K;
        float* dst = &sW[r * (K + 1) + cidx];
        dst[0] = v.x; dst[1] = v.y; dst[2] = v.z; dst[3] = v.w;
    }
    const float* xb = X + ((size_t)b * K << 16) + p0;
    for (int i = tid; i < K * QT; i += 256) {
        const int cidx = i / QT, po = (i % QT) * 4;
        const f4a v = *(const f4a*)(xb + ((size_t)cidx << 16) + po);
        float* dst = &sX[cidx * (NT + 1) + po];
        dst[0] = v.x; dst[1] = v.y; dst[2] = v.z; dst[3] = v.w;
    }
    __syncthreads();

    if constexpr (DO_LN) {
        if (tid < NT) {                           // per-pixel channel LN (biased var)
            float s = 0.f, s2 = 0.f;
            for (int c = 0; c < K; ++c) {
                float v = sX[c * (NT + 1) + tid];
                s += v; s2 += v * v;
            }
            float mu  = s  * (1.0f / K);
            float var = s2 * (1.0f / K) - mu * mu;
            sMu[tid] = mu;
            sRs[tid] = rsqrtf(var + 1e-6f);
        }
        __syncthreads();
        for (int i = tid; i < K * NT; i += 256) {
            int c = i / NT, p = i % NT;
            float v = sX[c * (NT + 1) + p];
            sX[c * (NT + 1) + p] = ln_w[c] * (v - sMu[p]) * sRs[p] + ln_b[c];
        }
        __syncthreads();
    }

    // Wave tile assignment.
    const int wv = tid >> 5, lane = tid & 31;
    const int mt = wv % MT, nt = wv / MT;
    const int m0 = mt * 16, q0 = nt * 16;
    const int hi = lane >> 4, lo = lane & 15;

    v8f acc = {};
    #pragma unroll
    for (int ks = 0; ks < K / 4; ++ks) {
        const int kk = ks * 4 + 2 * hi;           // ISA f32 A/B frag layout
        v2f a, bf;
        a.x  = sW[(m0 + lo) * (K + 1) + kk];
        a.y  = sW[(m0 + lo) * (K + 1) + kk + 1];
        bf.x = sX[kk       * (NT + 1) + q0 + lo];
        bf.y = sX[(kk + 1) * (NT + 1) + q0 + lo];
        acc = __builtin_amdgcn_wmma_f32_16x16x4_f32(
            false, a, false, bf, (short)0, acc, false, false);
    }

    // Epilogue + store (C/D layout: VGPR r -> M = m0 + r + 8*hi, N = q0+lo).
    const long pix = (long)p0 + q0 + lo;
    #pragma unroll
    for (int r = 0; r < 8; ++r) {
        const int m = m0 + r + 8 * hi;
        float v = acc[r] + bias[m];
        const size_t oidx = (((size_t)b * M + m) << 16) + pix;
        if constexpr (EPI == 1) {
            v = e0[oidx] * (e1[b * M + m] + v);   // x*(sca + attn)
        } else if constexpr (EPI == 2) {
            v = e0[oidx] + e1[m] * v;             // residual + beta/gamma scale
        }
        out[oidx] = v;
    }
}

// ---------------------------------------------------------------------------
// Depthwise KSxKS conv with dilation, "same" zero padding.
// 4 consecutive pixels per thread: interior taps load b128 (4B-aligned),
// borders fall back to guarded scalar taps. Output stores are aligned b128.
// ---------------------------------------------------------------------------
template <int KS>
__global__ __launch_bounds__(256) void dwconv4(
    const float* __restrict__ in, const float* __restrict__ w,
    const float* __restrict__ bias, float* __restrict__ out,
    int Cn, int dil, long nQuad)
{
    constexpr int R = KS / 2;
    for (long q = (long)blockIdx.x * blockDim.x + threadIdx.x; q < nQuad;
         q += (long)gridDim.x * blockDim.x) {
        const int  pq   = (int)(q & ((HW >> 2) - 1));
        const int  ww0  = (pq & (Ww / 4 - 1)) * 4;
        const int  hh   = pq >> 6;                 // Ww/4 == 64
        const long pl   = q >> 14;                 // b*Cn + c   (HW/4 == 1<<14)
        const int  c    = (int)(pl % Cn);
        const float* plane = in + (pl << 16);
        const float* wk = w + c * KS * KS;
        float a0, a1, a2, a3;
        a0 = a1 = a2 = a3 = bias[c];
        const bool fast = (ww0 - R * dil >= 0) && (ww0 + 3 + R * dil < Ww);
        #pragma unroll
        for (int kh = 0; kh < KS; ++kh) {
            const int ih = hh + (kh - R) * dil;
            if (ih < 0 || ih >= Hh) continue;
            const float* row = plane + ih * Ww;
            if (fast) {
                #pragma unroll
                for (int kw = 0; kw < KS; ++kw) {
                    const float wv = wk[kh * KS + kw];
                    const f4u v = *(const f4u*)(row + ww0 + (kw - R) * dil);
                    a0 += wv * v.x; a1 += wv * v.y; a2 += wv * v.z; a3 += wv * v.w;
                }
            } else {
                #pragma unroll
                for (int kw = 0; kw < KS; ++kw) {
                    const float wv = wk[kh * KS + kw];
                    const int base = ww0 + (kw - R) * dil;
                    if (base     >= 0 && base     < Ww) a0 += wv * row[base];
                    if (base + 1 >= 0 && base + 1 < Ww) a1 += wv * row[base + 1];
                    if (base + 2 >= 0 && base + 2 < Ww) a2 += wv * row[base + 2];
                    if (base + 3 >= 0 && base + 3 < Ww) a3 += wv * row[base + 3];
                }
            }
        }
        f4a o; o.x = a0; o.y = a1; o.z = a2; o.w = a3;
        *(f4a*)(out + (q << 2)) = o;
    }
}

// ---------------------------------------------------------------------------
// Fused tri-dilation depthwise(3x3, d=1/4/7) sum + SimpleGate (128 -> 64 ch).
// Same 4-pixel b128 scheme; max tap reach is 7, one fast-path test covers all.
// ---------------------------------------------------------------------------
__global__ __launch_bounds__(256) void mdgate4(
    const float* __restrict__ in,
    const float* __restrict__ wd1, const float* __restrict__ bd1,
    const float* __restrict__ wd4, const float* __restrict__ bd4,
    const float* __restrict__ wd7, const float* __restrict__ bd7,
    float* __restrict__ out, long nQuad)
{
    for (long q = (long)blockIdx.x * blockDim.x + threadIdx.x; q < nQuad;
         q += (long)gridDim.x * blockDim.x) {
        const int  pq  = (int)(q & ((HW >> 2) - 1));
        const int  ww0 = (pq & (Ww / 4 - 1)) * 4;
        const int  hh  = pq >> 6;
        const long pl  = q >> 14;                  // b*64 + c
        const int  c   = (int)(pl & 63);
        const int  b   = (int)(pl >> 6);
        const bool fast = (ww0 >= 7) && (ww0 + 3 + 7 < Ww);
        float g[2][4];
        #pragma unroll
        for (int s = 0; s < 2; ++s) {
            const int cc = c + 64 * s;
            const float* plane = in + ((long)(b * DWc + cc) << 16);
            const float binit = bd1[cc] + bd4[cc] + bd7[cc];
            float a0 = binit, a1 = binit, a2 = binit, a3 = binit;
            const float* wks[3]  = { wd1 + cc * 9, wd4 + cc * 9, wd7 + cc * 9 };
            const int    dils[3] = { 1, 4, 7 };
            #pragma unroll
            for (int d = 0; d < 3; ++d) {
                const int dil = dils[d];
                const float* wk = wks[d];
                #pragma unroll
                for (int kh = 0; kh < 3; ++kh) {
                    const int ih = hh + (kh - 1) * dil;
                    if (ih < 0 || ih >= Hh) continue;
                    const float* row = plane + ih * Ww;
                    if (fast) {
                        #pragma unroll
                        for (int kw = 0; kw < 3; ++kw) {
                            const float wv = wk[kh * 3 + kw];
                            const f4u v = *(const f4u*)(row + ww0 + (kw - 1) * dil);
                            a0 += wv * v.x; a1 += wv * v.y;
                            a2 += wv * v.z; a3 += wv * v.w;
                        }
                    } else {
                        #pragma unroll
                        for (int kw = 0; kw < 3; ++kw) {
                            const float wv = wk[kh * 3 + kw];
                            const int base = ww0 + (kw - 1) * dil;
                            if (base     >= 0 && base     < Ww) a0 += wv * row[base];
                            if (base + 1 >= 0 && base + 1 < Ww) a1 += wv * row[base + 1];
                            if (base + 2 >= 0 && base + 2 < Ww) a2 += wv * row[base + 2];
                            if (base + 3 >= 0 && base + 3 < Ww) a3 += wv * row[base + 3];
                        }
                    }
                }
            }
            g[s][0] = a0; g[s][1] = a1; g[s][2] = a2; g[s][3] = a3;
        }
        f4a o;
        o.x = g[0][0] * g[1][0]; o.y = g[0][1] * g[1][1];
        o.z = g[0][2] * g[1][2]; o.w = g[0][3] * g[1][3];
        *(f4a*)(out + (q << 2)) = o;
    }
}

// ---------------------------------------------------------------------------
// Global average pool: one block per (b,c) plane, b128 reads.
// ---------------------------------------------------------------------------
__global__ __launch_bounds__(256) void pool_kernel(
    const float* __restrict__ x, float* __restrict__ pooled)
{
    __shared__ float red[256];
    const int tid = threadIdx.x;
    const float* plane = x + ((size_t)blockIdx.x << 16);
    float s = 0.f;
    for (int i = tid; i < HW / 4; i += 256) {
        const f4a v = *(const f4a*)(plane + 4 * i);
        s += (v.x + v.y) + (v.z + v.w);
    }
    red[tid] = s;
    __syncthreads();
    for (int off = 128; off > 0; off >>= 1) {
        if (tid < off) red[tid] += red[tid + off];
        __syncthreads();
    }
    if (tid == 0) pooled[blockIdx.x] = red[0];
}

// SCA: sca[b][o] = sum_c sca_w[o][c]*mean[b][c] + sca_b[o]  (256 outputs, 1 blk)
__global__ __launch_bounds__(256) void sca_kernel(
    const float* __restrict__ pooled, const float* __restrict__ sw,
    const float* __restrict__ sb, float* __restrict__ sca)
{
    const int t = threadIdx.x;
    const int b = t >> 6, o = t & 63;
    float acc = 0.f;
    for (int c = 0; c < Cc; ++c) acc += sw[o * Cc + c] * pooled[b * Cc + c];
    sca[t] = acc * (1.0f / HW) + sb[o];
}

// GELU(exact) gate: out[c] = gelu(x[c]) * x[c+64], 128 -> 64 channels. b128 I/O.
__global__ __launch_bounds__(256) void gelugate4(
    const float* __restrict__ in, float* __restrict__ out, long nQuad)
{
    for (long q = (long)blockIdx.x * blockDim.x + threadIdx.x; q < nQuad;
         q += (long)gridDim.x * blockDim.x) {
        const int  p    = (int)(q & ((HW >> 2) - 1)) * 4;
        const long rest = q >> 14;
        const int  c    = (int)(rest & 63);
        const int  b    = (int)(rest >> 6);
        const float* base = in + ((long)(b * DWc) << 16);
        const f4a x1 = *(const f4a*)(base + ((long)c << 16) + p);
        const f4a x2 = *(const f4a*)(base + ((long)(c + 64) << 16) + p);
        f4a o;
        o.x = 0.5f * x1.x * (1.0f + erff(x1.x * 0.70710678118654752f)) * x2.x;
        o.y = 0.5f * x1.y * (1.0f + erff(x1.y * 0.70710678118654752f)) * x2.y;
        o.z = 0.5f * x1.z * (1.0f + erff(x1.z * 0.70710678118654752f)) * x2.z;
        o.w = 0.5f * x1.w * (1.0f + erff(x1.w * 0.70710678118654752f)) * x2.w;
        *(f4a*)(out + (q << 2)) = o;
    }
}

// ---------------------------------------------------------------------------
extern "C" void kernel_launch(void* const* d_in, const int* in_sizes, int n_in,
                              void* d_out, int out_size, void* d_ws, size_t ws_size,
                              hipStream_t stream)
{
    (void)in_sizes; (void)n_in; (void)out_size; (void)ws_size;
    const float* inp    = (const float*)d_in[0];
    const float* n1_w   = (const float*)d_in[1];
    const float* n1_b   = (const float*)d_in[2];
    const float* w1     = (const float*)d_in[3];
    const float* b1     = (const float*)d_in[4];
    const float* w2     = (const float*)d_in[5];
    const float* b2     = (const float*)d_in[6];
    const float* wd1    = (const float*)d_in[7];
    const float* bd1    = (const float*)d_in[8];
    const float* wd4    = (const float*)d_in[9];
    const float* bd4    = (const float*)d_in[10];
    const float* wd7    = (const float*)d_in[11];
    const float* bd7    = (const float*)d_in[12];
    const float* w3     = (const float*)d_in[13];
    const float* b3     = (const float*)d_in[14];
    const float* sca_w  = (const float*)d_in[15];
    const float* sca_b  = (const float*)d_in[16];
    const float* lka0_w = (const float*)d_in[17];
    const float* lka0_b = (const float*)d_in[18];
    const float* lkas_w = (const float*)d_in[19];
    const float* lkas_b = (const float*)d_in[20];
    const float* lka1_w = (const float*)d_in[21];
    const float* lka1_b = (const float*)d_in[22];
    const float* w4     = (const float*)d_in[23];
    const float* b4     = (const float*)d_in[24];
    const float* dw_w   = (const float*)d_in[25];
    const float* dw_b   = (const float*)d_in[26];
    const float* w5     = (const float*)d_in[27];
    const float* b5     = (const float*)d_in[28];
    const float* n2_w   = (const float*)d_in[29];
    const float* n2_b   = (const float*)d_in[30];
    const float* beta   = (const float*)d_in[31];
    const float* gamma  = (const float*)d_in[32];

    // Workspace carve (floats).
    const size_t nDW = (size_t)Bb * DWc * HW;   // 33,554,432
    const size_t nC  = (size_t)Bb * Cc  * HW;   // 16,777,216
    float* ws     = (float*)d_ws;
    float* bufA   = ws;                 // [B,128,HW]
    float* bufB   = bufA + nDW;         // [B,128,HW]
    float* bufC   = bufB + nDW;         // [B,64,HW]  (xg / gelu-gated)
    float* bufD   = bufC + nC;          // [B,64,HW]  (s = x*(sca+attn))
    float* bufY   = bufD + nC;          // [B,64,HW]  (y after first residual)
    float* pooled = bufY + nC;          // [B*C]
    float* scabuf = pooled + Bb * Cc;   // [B*C]

    const dim3 blk(256);
    const long qDW = (long)(nDW / 4);
    const long qC  = (long)(nC  / 4);
    const int gDW = (int)(qDW / 256);
    const int gC  = (int)(qC  / 256);

    // 1) LN1 + 1x1 conv 64->128 (WMMA f32)
    gemm1x1<128, 0, true><<<(int)(NPIX / 16), blk, 0, stream>>>(
        inp, w1, b1, n1_w, n1_b, nullptr, nullptr, bufA);
    // 2) depthwise 3x3
    dwconv4<3><<<gDW, blk, 0, stream>>>(bufA, w2, b2, bufB, DWc, 1, qDW);
    // 3) tri-dilation depthwise sum + SimpleGate -> xg
    mdgate4<<<gC, blk, 0, stream>>>(bufB, wd1, bd1, wd4, bd4, wd7, bd7, bufC, qC);
    // 4-5) SCA: pool + tiny GEMV
    pool_kernel<<<Bb * Cc, blk, 0, stream>>>(bufC, pooled);
    sca_kernel<<<1, blk, 0, stream>>>(pooled, sca_w, sca_b, scabuf);
    // 6) LKA depthwise 5x5
    dwconv4<5><<<gC, blk, 0, stream>>>(bufC, lka0_w, lka0_b, bufA, Cc, 1, qC);
    // 7) LKA depthwise 7x7 dilation 3
    dwconv4<7><<<gC, blk, 0, stream>>>(bufA, lkas_w, lkas_b, bufB, Cc, 3, qC);
    // 8) LKA 1x1 (WMMA) fused with  s = xg * (sca + attn)
    gemm1x1<64, 1, false><<<(int)(NPIX / 32), blk, 0, stream>>>(
        bufB, lka1_w, lka1_b, nullptr, nullptr, bufC, scabuf, bufD);
    // 9) w3 1x1 (WMMA) fused with  y = inp + beta * (.)
    gemm1x1<64, 2, false><<<(int)(NPIX / 32), blk, 0, stream>>>(
        bufD, w3, b3, nullptr, nullptr, inp, beta, bufY);
    // 10) LN2 + w4 1x1 64->128 (WMMA)
    gemm1x1<128, 0, true><<<(int)(NPIX / 16), blk, 0, stream>>>(
        bufY, w4, b4, n2_w, n2_b, nullptr, nullptr, bufA);
    // 11) FFN depthwise 3x3
    dwconv4<3><<<gDW, blk, 0, stream>>>(bufA, dw_w, dw_b, bufB, DWc, 1, qDW);
    // 12) GELU gate 128 -> 64
    gelugate4<<<gC, blk, 0, stream>>>(bufB, bufC, qC);
    // 13) w5 1x1 (WMMA) fused with  out = y + gamma * (.)
    gemm1x1<64, 2, false><<<(int)(NPIX / 32), blk, 0, stream>>>(
        bufC, w5, b5, nullptr, nullptr, bufY, gamma, (float*)d_out);
}